// Adjacency_20177756357157
// MI455X (gfx1250) — compile-verified
//
#include <hip/hip_runtime.h>
#include <hip/hip_bf16.h>

// ---------------------------------------------------------------------------
// Problem constants (from reference): N=64 nodes, F=256 features, U=N*N=4096.
// out = relu(relu(a_vec @ W1) @ W2) for three independent (adj, W1, W2) sets.
// Memory-bound: 6 x 64MB fp32 weights streamed once -> ~17us floor @ 23.3TB/s.
// ---------------------------------------------------------------------------
#define N_NODES 64
#define N_FEAT  256
#define U       4096          // flattened vector length == GEMV K == GEMV N
#define KSPLIT  8             // K-dimension split for occupancy
#define KSEG    (U / KSPLIT)  // 512 k-rows per wave
#define NTILES  (U / 16)      // 256 column tiles of 16
#define GEMV_WAVES (3 * KSPLIT * NTILES)   // 6144 waves
#define GEMV_BLOCKS (GEMV_WAVES / 8)       // 768 blocks of 256 threads

typedef __attribute__((ext_vector_type(2))) float v2f;
typedef __attribute__((ext_vector_type(8))) float v8f;

// ---------------------------------------------------------------------------
// Kernel 1: masked pairwise-distance matrices.
// One block per adjacency. node_vec staged transposed in LDS [f][node] so the
// inner loop reads are broadcast (i) and stride-1 (j): conflict-free.
// ---------------------------------------------------------------------------
__global__ __launch_bounds__(256) void dist_kernel(
    const float* __restrict__ adj0, const float* __restrict__ adj1,
    const float* __restrict__ adj2, const float* __restrict__ nv,
    float* __restrict__ aVec) {
  __shared__ float snv[N_FEAT * N_NODES];  // 64 KB LDS, transposed [f][node]
  const int g = blockIdx.x;
  const int tid = threadIdx.x;
  for (int idx = tid; idx < N_NODES * N_FEAT; idx += 256) {
    int node = idx >> 8;       // nv is [node][f] row-major
    int f = idx & 255;
    snv[f * N_NODES + node] = nv[idx];
  }
  __syncthreads();
  const float* adj = (g == 0) ? adj0 : (g == 1) ? adj1 : adj2;
  for (int p = tid; p < N_NODES * N_NODES; p += 256) {
    int i = p >> 6;
    int j = p & 63;
    float acc = 0.f;
#pragma unroll 4
    for (int f = 0; f < N_FEAT; ++f) {
      float d = snv[f * N_NODES + i] - snv[f * N_NODES + j];
      acc = fmaf(d, d, acc);
    }
    float m = adj[p];
    aVec[g * U + p] = (m == 1.0f) ? sqrtf(acc) : 0.f;
  }
}

// ---------------------------------------------------------------------------
// Kernel 2: fp32 GEMV partials via V_WMMA_F32_16X16X4_F32.
//   h[n] = sum_k v[k] * W[k*U + n]
// Per wave: one 16-column tile, one K segment of 512 rows, 128 WMMAs.
//   A[m][k'] = v[k0+k']   (broadcast over m; per documented A 16x4 layout:
//                          VGPR0 = {K0 | K2}, VGPR1 = {K1 | K3} lane halves)
//   B[k'][n] = W[k0+k'][n_base+n]  (row-major tile; VGPR0 = rows k0,k0+2,
//                                   VGPR1 = rows k0+1,k0+3 — coalesced 64B
//                                   segments per half-wave)
// D rows are all identical; lanes 0..15 of acc[0] hold the 16 partials.
// Partials written to workspace (deterministic tree-reduce afterwards).
// ---------------------------------------------------------------------------
__global__ __launch_bounds__(256) void gemv_wmma_kernel(
    const float* __restrict__ v,   // [3][U], g-major
    const float* __restrict__ w0, const float* __restrict__ w1,
    const float* __restrict__ w2,
    float* __restrict__ part) {    // [3][KSPLIT][U]
  const int wave = (blockIdx.x * blockDim.x + threadIdx.x) >> 5;
  const int lane = threadIdx.x & 31;

  const int g   = wave >> 11;        // / (KSPLIT*NTILES) = /2048
  const int rem = wave & 2047;
  const int ks  = rem >> 8;          // / NTILES = /256
  const int nt  = rem & 255;
  const int n_base = nt << 4;

  const float* __restrict__ W  = (g == 0) ? w0 : (g == 1) ? w1 : w2;
  const float* __restrict__ vp = v + g * U + ks * KSEG;

  const int nOff = lane & 15;          // column within tile
  const int kOff = (lane >> 4) << 1;   // 0 for lanes 0-15, 2 for lanes 16-31
  const float* __restrict__ wp = W + (size_t)(ks * KSEG + kOff) * U + n_base + nOff;

  v8f acc = {};
#pragma unroll 4
  for (int k = 0; k < KSEG; k += 4) {
    // B operand: two coalesced row loads (rows k+kOff and k+kOff+1)
    float b0 = wp[(size_t)k * U];
    float b1 = wp[(size_t)(k + 1) * U];
    // A operand: vector broadcast, lane-half selects K pair
    float a0 = vp[k + kOff];
    float a1 = vp[k + kOff + 1];
    v2f A = {a0, a1};
    v2f B = {b0, b1};
    acc = __builtin_amdgcn_wmma_f32_16x16x4_f32(
        /*neg_a=*/false, A, /*neg_b=*/false, B,
        /*c_mod=*/(short)0, acc, /*reuse_a=*/false, /*reuse_b=*/false);
  }
  if (lane < 16) {
    part[((size_t)g * KSPLIT + ks) * U + n_base + lane] = acc[0];
  }
}

// ---------------------------------------------------------------------------
// Kernel 3: deterministic reduce over KSPLIT partials + ReLU.
// ---------------------------------------------------------------------------
__global__ __launch_bounds__(256) void reduce_relu_kernel(
    const float* __restrict__ part,  // [3][KSPLIT][U]
    float* __restrict__ out) {       // [3][U]
  const int t = blockIdx.x * blockDim.x + threadIdx.x;
  if (t >= 3 * U) return;
  const int g = t >> 12;
  const int n = t & (U - 1);
  const float* p = part + (size_t)g * KSPLIT * U + n;
  float s = 0.f;
#pragma unroll
  for (int i = 0; i < KSPLIT; ++i) s += p[i * U];
  out[t] = fmaxf(s, 0.f);
}

// ---------------------------------------------------------------------------
// Host-side launch sequence (single stream, graph-capture safe).
// Workspace layout (floats):
//   aVec [3*U]            @ 0
//   p1   [3*KSPLIT*U]     @ 12288
//   r1   [3*U]            @ 110592
//   p2   [3*KSPLIT*U]     @ 122880     total ~865 KB
// ---------------------------------------------------------------------------
extern "C" void kernel_launch(void* const* d_in, const int* in_sizes, int n_in,
                              void* d_out, int out_size, void* d_ws, size_t ws_size,
                              hipStream_t stream) {
  const float* adj0 = (const float*)d_in[0];
  const float* adj1 = (const float*)d_in[1];
  const float* adj2 = (const float*)d_in[2];
  const float* nv   = (const float*)d_in[3];
  const float* w0_1 = (const float*)d_in[4];
  const float* w0_2 = (const float*)d_in[5];
  const float* w1_1 = (const float*)d_in[6];
  const float* w1_2 = (const float*)d_in[7];
  const float* w2_1 = (const float*)d_in[8];
  const float* w2_2 = (const float*)d_in[9];
  float* out = (float*)d_out;  // [out0 | out1 | out2], each 64*64 fp32

  float* ws   = (float*)d_ws;
  float* aVec = ws;                       // 3*U
  float* p1   = aVec + 3 * U;             // 3*KSPLIT*U
  float* r1   = p1 + 3 * KSPLIT * U;      // 3*U
  float* p2   = r1 + 3 * U;               // 3*KSPLIT*U

  // 1) masked distance matrices -> aVec
  dist_kernel<<<3, 256, 0, stream>>>(adj0, adj1, adj2, nv, aVec);
  // 2) layer-1 GEMV partials (streams 3 x 64MB of W1 weights)
  gemv_wmma_kernel<<<GEMV_BLOCKS, 256, 0, stream>>>(aVec, w0_1, w1_1, w2_1, p1);
  // 3) reduce + ReLU -> r1
  reduce_relu_kernel<<<(3 * U + 255) / 256, 256, 0, stream>>>(p1, r1);
  // 4) layer-2 GEMV partials (streams 3 x 64MB of W2 weights)
  gemv_wmma_kernel<<<GEMV_BLOCKS, 256, 0, stream>>>(r1, w0_2, w1_2, w2_2, p2);
  // 5) reduce + ReLU -> final outputs
  reduce_relu_kernel<<<(3 * U + 255) / 256, 256, 0, stream>>>(p2, out);
}